// EnergyPool2d_15470472200686
// MI455X (gfx1250) — compile-verified
//
#include <hip/hip_runtime.h>
#include <hip/hip_bf16.h>
#include <stdint.h>

// Problem constants (from reference): N=16, C=64, H=W=128, 3x3 windows stride 1.
#define PLANES   1024          // N*C
#define Hdim     128
#define Wdim     128
#define HW       16384         // 128*128
#define WO       126           // output window grid per side
#define NWIN     15876         // 126*126
#define CHUNKS   4096          // HW floats / 4 floats per b128 = 4096 16B chunks
#define BLOCK    256

__global__ __launch_bounds__(BLOCK)
void EnergyPool2d_kernel(const float* __restrict__ x, float* __restrict__ out) {
    // 64 KB input tile + 64 KB counts tile = 128 KB LDS (<= 320 KB per WGP, 2 WGs/WGP)
    __shared__ __align__(16) float xs[HW];
    __shared__ __align__(16) float cnt[HW];

    const int tid   = threadIdx.x;
    const int plane = blockIdx.x;

    const float* gx = x   + (size_t)plane * HW;
    float*       go = out + (size_t)plane * HW;

    // Wave-relative LDS byte addresses: low 32 bits of the flat pointer are the
    // LDS offset.
    const uint32_t xs_base  = (uint32_t)(uintptr_t)&xs[0];
    const uint32_t cnt_base = (uint32_t)(uintptr_t)&cnt[0];

    // ---- Phase 1: async DMA global -> LDS (input plane), zero counts in LDS ----
    #pragma unroll 4
    for (int i = tid; i < CHUNKS; i += BLOCK) {
        uint32_t lds_addr = xs_base + (uint32_t)i * 16u;
        uint64_t gaddr    = (uint64_t)(uintptr_t)(gx + i * 4);
        // per-lane 16B async copy to LDS, tracked by ASYNCcnt
        asm volatile("global_load_async_to_lds_b128 %0, %1, off"
                     :: "v"(lds_addr), "v"(gaddr) : "memory");
    }
    {
        float4 z; z.x = 0.f; z.y = 0.f; z.z = 0.f; z.w = 0.f;
        float4* c4 = (float4*)cnt;
        #pragma unroll 4
        for (int i = tid; i < CHUNKS; i += BLOCK) {
            c4[i] = z;                       // ds_store_b128
        }
    }
    asm volatile("s_wait_asynccnt 0" ::: "memory");
    __syncthreads();

    // ---- Phase 2: per-window argmax/argmin in LDS, scatter via LDS float atomics ----
    // window (wi,wj): flat element (wi+a)*128 + (wj+b); loop order matches
    // jnp.argmax/argmin flattened order; strict compares keep first occurrence.
    for (int w = tid; w < NWIN; w += BLOCK) {
        const int wi   = w / WO;
        const int wj   = w - wi * WO;
        const int base = wi * Wdim + wj;

        float v0 = xs[base];
        float bmax = v0; int imax = base;
        float bmin = v0; int imin = base;

        #pragma unroll
        for (int a = 0; a < 3; ++a) {
            #pragma unroll
            for (int b = 0; b < 3; ++b) {
                if (a == 0 && b == 0) continue;
                const int idx = base + a * Wdim + b;
                const float v = xs[idx];
                if (v > bmax) { bmax = v; imax = idx; }
                if (v < bmin) { bmin = v; imin = idx; }
            }
        }
        atomicAdd(&cnt[imax],  1.0f);        // ds_add_f32
        atomicAdd(&cnt[imin], -1.0f);        // ds_add_f32
    }
    __syncthreads();

    // ---- Phase 3: async DMA LDS -> global (counts plane) ----
    #pragma unroll 4
    for (int i = tid; i < CHUNKS; i += BLOCK) {
        uint32_t lds_addr = cnt_base + (uint32_t)i * 16u;
        uint64_t gaddr    = (uint64_t)(uintptr_t)(go + i * 4);
        asm volatile("global_store_async_from_lds_b128 %0, %1, off"
                     :: "v"(gaddr), "v"(lds_addr) : "memory");
    }
    asm volatile("s_wait_asynccnt 0" ::: "memory");
}

extern "C" void kernel_launch(void* const* d_in, const int* in_sizes, int n_in,
                              void* d_out, int out_size, void* d_ws, size_t ws_size,
                              hipStream_t stream) {
    const float* x = (const float*)d_in[0];
    float* out     = (float*)d_out;
    (void)in_sizes; (void)n_in; (void)out_size; (void)d_ws; (void)ws_size;
    EnergyPool2d_kernel<<<PLANES, BLOCK, 0, stream>>>(x, out);
}